// GCN_71640054497481
// MI455X (gfx1250) — compile-verified
//
#include <hip/hip_runtime.h>

#define HIDF 64

typedef __attribute__((ext_vector_type(2))) float v2f;
typedef __attribute__((ext_vector_type(8))) float v8f;

static inline int cdiv(long long a, long long b) { return (int)((a + b - 1) / b); }

// ---------------------------------------------------------------------------
// Utility: zero a float region (agg buffers, degree, pooling accumulators)
// ---------------------------------------------------------------------------
__global__ void gcn_zero_f32(float* __restrict__ p, long long n) {
  long long i = (long long)blockIdx.x * blockDim.x + threadIdx.x;
  if (i < n) p[i] = 0.0f;
}

// ---------------------------------------------------------------------------
// Degree at dst (+1 self loop added later), then dinv = rsqrt(deg+1)
// ---------------------------------------------------------------------------
__global__ void gcn_deg_accum(const int* __restrict__ dst, float* __restrict__ deg, int E) {
  int e = blockIdx.x * blockDim.x + threadIdx.x;
  if (e < E) atomicAdd(&deg[dst[e]], 1.0f);
}

__global__ void gcn_make_dinv(float* __restrict__ deg, int N) {
  int i = blockIdx.x * blockDim.x + threadIdx.x;
  if (i < N) deg[i] = rsqrtf(deg[i] + 1.0f);
}

__global__ void gcn_make_norm(const int* __restrict__ src, const int* __restrict__ dst,
                              const float* __restrict__ dinv, float* __restrict__ norm, int E) {
  int e = blockIdx.x * blockDim.x + threadIdx.x;
  if (e < E) norm[e] = dinv[src[e]] * dinv[dst[e]];
}

// ---------------------------------------------------------------------------
// C[M x 64] = A[M x K] * W[K x 64]   (row-major, fp32, WMMA 16x16x4_f32)
//
// W is staged into LDS PRE-SWIZZLED so every B fragment is one aligned
// ds_load_b64:   Wlds[(k/2)*128 + 2n + {0,1}] = W[k + {0,1}][n]
// A lane (half hh = lane>>4, col/row r = lane&15) reads pair-group
// (k/2 + hh), giving rows k+2*hh, k+2*hh+1 as required by the B layout.
//
// One wave computes a 32-row x 64-col strip: 8 C tiles; each B fragment set
// feeds 8 independent WMMAs (no D->A/B hazards back-to-back).
//   A frag (16x4): lane = (hh<<4)|row ; v2f = A[row][k + 2*hh + {0,1}]
//   C/D (16x16):   VGPR i = row (hh*8 + i), col = lane&15
// ---------------------------------------------------------------------------
__global__ __launch_bounds__(128) void gcn_gemm_wmma_f32(
    const float* __restrict__ A, const float* __restrict__ W,
    float* __restrict__ C, int M, int K) {
  __shared__ float Wlds[64 * 128];  // (K/2 <= 64) pair-groups x 128 floats = 32 KB

  const int tid = threadIdx.x;
  // cooperative swizzled stage of W[K x 64] into LDS
  {
    const int npair = (K >> 1) * HIDF;  // K/2 * 64 element pairs
    for (int i = tid; i < npair; i += 128) {
      int kk = i >> 6;  // pair-group (k = 2*kk)
      int n  = i & 63;  // column
      float2 v;
      v.x = W[(size_t)(2 * kk) * HIDF + n];
      v.y = W[(size_t)(2 * kk + 1) * HIDF + n];
      *(float2*)(Wlds + (size_t)kk * 128 + 2 * n) = v;  // ds_store_b64
    }
  }
  __syncthreads();

  const int lane = tid & 31;
  const int wave = tid >> 5;
  const int m0   = (blockIdx.x * 4 + wave) * 32;  // 32 rows per wave

  const int r  = lane & 15;        // row (A) / col (B,C)
  const int hh = lane >> 4;        // lane half -> K sub-offset 2*hh
  const int kb = hh * 2;

  const int row0 = m0 + r;
  const int row1 = m0 + 16 + r;
  const float am0 = (row0 < M) ? 1.0f : 0.0f;  // branch-free OOB masking
  const float am1 = (row1 < M) ? 1.0f : 0.0f;
  const float* __restrict__ A0 = A + (size_t)(row0 < M ? row0 : 0) * K + kb;
  const float* __restrict__ A1 = A + (size_t)(row1 < M ? row1 : 0) * K + kb;
  const float* Wl = Wlds + hh * 128 + 2 * r;  // lane's base within pair-group

  v8f c00 = {}, c01 = {}, c02 = {}, c03 = {};
  v8f c10 = {}, c11 = {}, c12 = {}, c13 = {};

#pragma unroll 4
  for (int k = 0; k < K; k += 4) {
    const float2 a0v = *(const float2*)(A0 + k);  // fused global b64 load
    const float2 a1v = *(const float2*)(A1 + k);
    v2f a0; a0.x = a0v.x * am0; a0.y = a0v.y * am0;
    v2f a1; a1.x = a1v.x * am1; a1.y = a1v.y * am1;

    const float* wk = Wl + (k >> 1) * 128;        // pair-group k/2 (+hh)
    const v2f b0 = *(const v2f*)(wk + 0);         // ds_load_b64 each
    const v2f b1 = *(const v2f*)(wk + 32);
    const v2f b2 = *(const v2f*)(wk + 64);
    const v2f b3 = *(const v2f*)(wk + 96);

    c00 = __builtin_amdgcn_wmma_f32_16x16x4_f32(false, a0, false, b0, (short)0, c00, false, false);
    c01 = __builtin_amdgcn_wmma_f32_16x16x4_f32(false, a0, false, b1, (short)0, c01, false, false);
    c02 = __builtin_amdgcn_wmma_f32_16x16x4_f32(false, a0, false, b2, (short)0, c02, false, false);
    c03 = __builtin_amdgcn_wmma_f32_16x16x4_f32(false, a0, false, b3, (short)0, c03, false, false);
    c10 = __builtin_amdgcn_wmma_f32_16x16x4_f32(false, a1, false, b0, (short)0, c10, false, false);
    c11 = __builtin_amdgcn_wmma_f32_16x16x4_f32(false, a1, false, b1, (short)0, c11, false, false);
    c12 = __builtin_amdgcn_wmma_f32_16x16x4_f32(false, a1, false, b2, (short)0, c12, false, false);
    c13 = __builtin_amdgcn_wmma_f32_16x16x4_f32(false, a1, false, b3, (short)0, c13, false, false);
  }

  const int rowOff = hh * 8;
  const int col    = r;
#pragma unroll
  for (int i = 0; i < 8; ++i) {
    int mm = m0 + rowOff + i;
    if (mm < M) {
      float* Crow = C + (size_t)mm * HIDF + col;
      Crow[0]  = c00[i];
      Crow[16] = c01[i];
      Crow[32] = c02[i];
      Crow[48] = c03[i];
    }
    int mm1 = mm + 16;
    if (mm1 < M) {
      float* Crow = C + (size_t)mm1 * HIDF + col;
      Crow[0]  = c10[i];
      Crow[16] = c11[i];
      Crow[32] = c12[i];
      Crow[48] = c13[i];
    }
  }
}

// ---------------------------------------------------------------------------
// Edge scatter: agg[dst] += h[src] * norm.  16 threads/edge, float4 each.
// ---------------------------------------------------------------------------
__global__ void gcn_scatter(const float* __restrict__ h, const float* __restrict__ norm,
                            const int* __restrict__ src, const int* __restrict__ dst,
                            float* __restrict__ agg, int E) {
  long long t = (long long)blockIdx.x * blockDim.x + threadIdx.x;
  long long e = t >> 4;
  if (e >= E) return;
  int c = (int)(t & 15) * 4;
  int s = src[e], d = dst[e];
  float n = norm[e];
  const float4 hv = *(const float4*)(h + (size_t)s * HIDF + c);
  float* ap = agg + (size_t)d * HIDF + c;
  atomicAdd(ap + 0, hv.x * n);
  atomicAdd(ap + 1, hv.y * n);
  atomicAdd(ap + 2, hv.z * n);
  atomicAdd(ap + 3, hv.w * n);
}

// ---------------------------------------------------------------------------
// Epilogue (in place): h = relu(agg + h*dinv^2 + bias)
// ---------------------------------------------------------------------------
__global__ void gcn_epilogue(float* __restrict__ h, const float* __restrict__ agg,
                             const float* __restrict__ dinv, const float* __restrict__ bias,
                             int N) {
  long long t = (long long)blockIdx.x * blockDim.x + threadIdx.x;
  long long i = t >> 4;
  if (i >= N) return;
  int c = (int)(t & 15) * 4;
  float s = dinv[i]; s *= s;
  float4 hv = *(float4*)(h + (size_t)i * HIDF + c);
  float4 av = *(const float4*)(agg + (size_t)i * HIDF + c);
  const float4 bv = *(const float4*)(bias + c);
  float4 o;
  o.x = fmaxf(av.x + hv.x * s + bv.x, 0.0f);
  o.y = fmaxf(av.y + hv.y * s + bv.y, 0.0f);
  o.z = fmaxf(av.z + hv.z * s + bv.z, 0.0f);
  o.w = fmaxf(av.w + hv.w * s + bv.w, 0.0f);
  *(float4*)(h + (size_t)i * HIDF + c) = o;
}

// ---------------------------------------------------------------------------
// Global mean pool (accumulate): sums[g] += h[i], cnts[g] += 1
// ---------------------------------------------------------------------------
__global__ void gcn_pool(const float* __restrict__ h, const int* __restrict__ batch,
                         float* __restrict__ sums, float* __restrict__ cnts, int N) {
  long long t = (long long)blockIdx.x * blockDim.x + threadIdx.x;
  long long i = t >> 4;
  if (i >= N) return;
  int c = (int)(t & 15) * 4;
  int g = batch[i];
  const float4 hv = *(const float4*)(h + (size_t)i * HIDF + c);
  float* sp = sums + (size_t)g * HIDF + c;
  atomicAdd(sp + 0, hv.x);
  atomicAdd(sp + 1, hv.y);
  atomicAdd(sp + 2, hv.z);
  atomicAdd(sp + 3, hv.w);
  if (c == 0) atomicAdd(&cnts[g], 1.0f);
}

// ---------------------------------------------------------------------------
// Head: out[g] = (sums[g]/max(cnt,1)) . Wlin + blin     (G = 64 threads)
// ---------------------------------------------------------------------------
__global__ void gcn_head(const float* __restrict__ sums, const float* __restrict__ cnts,
                         const float* __restrict__ Wlin, const float* __restrict__ blin,
                         float* __restrict__ out, int G) {
  int g = blockIdx.x * blockDim.x + threadIdx.x;
  if (g >= G) return;
  float inv = 1.0f / fmaxf(cnts[g], 1.0f);
  float acc = 0.0f;
#pragma unroll
  for (int f = 0; f < HIDF; ++f) acc += sums[(size_t)g * HIDF + f] * inv * Wlin[f];
  out[g] = acc + blin[0];
}

// ---------------------------------------------------------------------------
extern "C" void kernel_launch(void* const* d_in, const int* in_sizes, int n_in,
                              void* d_out, int out_size, void* d_ws, size_t ws_size,
                              hipStream_t stream) {
  const float* x     = (const float*)d_in[0];  // [N, 128]
  const int*   eidx  = (const int*)d_in[1];    // [2, E]
  const int*   batch = (const int*)d_in[2];    // [N]
  const float* W1    = (const float*)d_in[3];  // [128, 64]
  const float* b1    = (const float*)d_in[4];  // [64]
  const float* W2    = (const float*)d_in[5];  // [64, 64]
  const float* b2    = (const float*)d_in[6];  // [64]
  const float* Wlin  = (const float*)d_in[7];  // [64, 1]
  const float* blin  = (const float*)d_in[8];  // [1]
  float* out = (float*)d_out;

  const int E  = in_sizes[1] / 2;
  const int N  = in_sizes[2];
  const int K1 = in_sizes[0] / N;  // 128
  const int G  = out_size;         // 64 graphs

  const int* src = eidx;
  const int* dst = eidx + E;

  // workspace layout (floats)
  float* ws   = (float*)d_ws;
  float* dinv = ws;                       // [N]   (degree, then rsqrt)
  float* norm = dinv + N;                 // [E]
  float* h1   = norm + E;                 // [N*64]
  float* agg  = h1 + (size_t)N * HIDF;    // [N*64]
  float* h2   = agg + (size_t)N * HIDF;   // [N*64]
  float* sums = h2 + (size_t)N * HIDF;    // [G*64]
  float* cnts = sums + (size_t)G * HIDF;  // [G]

  const int TB = 256;
  const long long NF = (long long)N * HIDF;

  // --- normalization (shared by both layers) ---
  gcn_zero_f32<<<cdiv(N, TB), TB, 0, stream>>>(dinv, N);
  gcn_zero_f32<<<cdiv((long long)G * HIDF + G, TB), TB, 0, stream>>>(sums, (long long)G * HIDF + G);
  gcn_deg_accum<<<cdiv(E, TB), TB, 0, stream>>>(dst, dinv, E);
  gcn_make_dinv<<<cdiv(N, TB), TB, 0, stream>>>(dinv, N);
  gcn_make_norm<<<cdiv(E, TB), TB, 0, stream>>>(src, dst, dinv, norm, E);

  // --- layer 1 ---
  gcn_gemm_wmma_f32<<<cdiv((long long)N, 128), 128, 0, stream>>>(x, W1, h1, N, K1);
  gcn_zero_f32<<<cdiv(NF, TB), TB, 0, stream>>>(agg, NF);
  gcn_scatter<<<cdiv((long long)E * 16, TB), TB, 0, stream>>>(h1, norm, src, dst, agg, E);
  gcn_epilogue<<<cdiv((long long)N * 16, TB), TB, 0, stream>>>(h1, agg, dinv, b1, N);

  // --- layer 2 ---
  gcn_gemm_wmma_f32<<<cdiv((long long)N, 128), 128, 0, stream>>>(h1, W2, h2, N, HIDF);
  gcn_zero_f32<<<cdiv(NF, TB), TB, 0, stream>>>(agg, NF);
  gcn_scatter<<<cdiv((long long)E * 16, TB), TB, 0, stream>>>(h2, norm, src, dst, agg, E);
  gcn_epilogue<<<cdiv((long long)N * 16, TB), TB, 0, stream>>>(h2, agg, dinv, b2, N);

  // --- pool + head ---
  gcn_pool<<<cdiv((long long)N * 16, TB), TB, 0, stream>>>(h2, batch, sums, cnts, N);
  gcn_head<<<1, 64, 0, stream>>>(sums, cnts, Wlin, blin, out, G);
}